// Net_68101001445971
// MI455X (gfx1250) — compile-verified
//
#include <hip/hip_runtime.h>
#include <math.h>

// ---- static problem constants (match reference) ----
static constexpr int Bsz  = 16;
static constexpr int P    = 1024;
static constexpr int KNN  = 20;
static constexpr int F1   = 64;
static constexpr int C2   = 128;
static constexpr int NCLS = 40;
static constexpr int Npts = Bsz * P;     // 16384
static constexpr int NC   = Npts / 2;    // 8192
static constexpr int PC   = P / 2;       // 512
static constexpr int FDIM = F1 * 3;      // 192
static constexpr int EDGES = NC * KNN;   // 163840
#define BN_EPS 1e-5f

typedef float v2f __attribute__((ext_vector_type(2)));
typedef float v8f __attribute__((ext_vector_type(8)));

// ============================================================
// Stage 1: kNN on xyz positions (per batch of 1024 points)
// One thread per point; candidate positions tiled through LDS.
// ============================================================
__global__ void knn1_kernel(const float* __restrict__ pos, int* __restrict__ nbr) {
  const int g = blockIdx.x * 256 + threadIdx.x;      // 0..16383
  const int base = (g >> 10) << 10;                  // batch base
  const float px = pos[g * 3 + 0];
  const float py = pos[g * 3 + 1];
  const float pz = pos[g * 3 + 2];
  float bd[KNN]; int bi[KNN];
#pragma unroll
  for (int k = 0; k < KNN; ++k) { bd[k] = 3.0e38f; bi[k] = 0; }
  __shared__ float sx[256], sy[256], sz[256];
  for (int t = 0; t < P; t += 256) {
    const int j = base + t + threadIdx.x;
    sx[threadIdx.x] = pos[j * 3 + 0];
    sy[threadIdx.x] = pos[j * 3 + 1];
    sz[threadIdx.x] = pos[j * 3 + 2];
    __syncthreads();
    for (int jj = 0; jj < 256; ++jj) {
      const int cand = base + t + jj;
      const float dx = sx[jj] - px, dy = sy[jj] - py, dz = sz[jj] - pz;
      const float d2 = dx * dx + dy * dy + dz * dz;
      if (cand != g && d2 < bd[KNN - 1]) {
        int p = KNN - 1;
        while (p > 0 && bd[p - 1] > d2) { bd[p] = bd[p - 1]; bi[p] = bi[p - 1]; --p; }
        bd[p] = d2; bi[p] = cand;
      }
    }
    __syncthreads();
  }
  for (int k = 0; k < KNN; ++k) nbr[g * KNN + k] = bi[k];
}

// ============================================================
// Stage 2: DS directional edge features, max over K neighbors.
// Block = one point, 64 threads = 64 filters.
// ============================================================
__global__ void ds_kernel(const float* __restrict__ pos, const int* __restrict__ nbr,
                          const float* __restrict__ Wds, const float* __restrict__ bds,
                          float* __restrict__ fdd) {
  const int i = blockIdx.x;       // point
  const int f = threadIdx.x;      // filter
  __shared__ float rel[KNN][3];
  __shared__ float inv[KNN];
  __shared__ float own[3];
  if (f < 3) own[f] = pos[i * 3 + f];
  __syncthreads();
  if (f < KNN) {
    const int j = nbr[i * KNN + f];
    const float rx = pos[j * 3 + 0] - own[0];
    const float ry = pos[j * 3 + 1] - own[1];
    const float rz = pos[j * 3 + 2] - own[2];
    rel[f][0] = rx; rel[f][1] = ry; rel[f][2] = rz;
    inv[f] = 1.0f / (sqrtf(rx * rx + ry * ry + rz * rz) + 1e-8f);
  }
  __syncthreads();
  const float w0 = Wds[f], w1 = Wds[F1 + f], w2 = Wds[2 * F1 + f], bf = bds[f];
  float m0 = -3.0e38f, m1 = -3.0e38f, m2 = -3.0e38f;
  for (int k = 0; k < KNN; ++k) {
    const float rx = rel[k][0], ry = rel[k][1], rz = rel[k][2];
    const float h = fmaxf(rx * w0 + ry * w1 + rz * w2 + bf, 0.0f);
    const float s = h * inv[k];
    m0 = fmaxf(m0, s * rx);
    m1 = fmaxf(m1, s * ry);
    m2 = fmaxf(m2, s * rz);
  }
  fdd[(size_t)i * FDIM + f * 3 + 0] = m0;
  fdd[(size_t)i * FDIM + f * 3 + 1] = m1;
  fdd[(size_t)i * FDIM + f * 3 + 2] = m2;
}

// ============================================================
// Stage 3: graclus pair-pool (max of consecutive points) + per-channel norms.
// ============================================================
__global__ void pool_kernel(const float* __restrict__ fdd, float* __restrict__ fgra,
                            float* __restrict__ fn) {
  const int idx = blockIdx.x * 256 + threadIdx.x;
  if (idx >= NC * F1) return;
  const int cI = idx / F1, f = idx % F1;
  const float* a = fdd + (size_t)(2 * cI) * FDIM + f * 3;
  const float* b = a + FDIM;
  const float x = fmaxf(a[0], b[0]);
  const float y = fmaxf(a[1], b[1]);
  const float z = fmaxf(a[2], b[2]);
  float* o = fgra + (size_t)cI * FDIM + f * 3;
  o[0] = x; o[1] = y; o[2] = z;
  fn[cI * F1 + f] = sqrtf(x * x + y * y + z * z);
}

// ============================================================
// Stage 4: kNN on 64-dim feature norms (per batch of 512 clusters).
// Rank by (|fj|^2 - 2 fi.fj) which is order-equivalent to ||fi-fj||^2.
// ============================================================
__global__ void knn2_kernel(const float* __restrict__ fn, int* __restrict__ nbr2) {
  const int g = blockIdx.x * 256 + threadIdx.x;     // 0..8191
  const int base = (g >> 9) << 9;                   // batch base (512 clusters)
  float f[F1];
#pragma unroll
  for (int d = 0; d < F1; ++d) f[d] = fn[(size_t)g * F1 + d];
  float bd[KNN]; int bi[KNN];
#pragma unroll
  for (int k = 0; k < KNN; ++k) { bd[k] = 3.0e38f; bi[k] = 0; }
  __shared__ float sf[64][F1 + 1];
  __shared__ float ssq[64];
  for (int t = 0; t < PC; t += 64) {
    for (int e = threadIdx.x; e < 64 * F1; e += 256) {
      const int cj = e >> 6, d = e & 63;
      sf[cj][d] = fn[(size_t)(base + t + cj) * F1 + d];
    }
    __syncthreads();
    if (threadIdx.x < 64) {
      float s = 0.0f;
      for (int d = 0; d < F1; ++d) { const float v = sf[threadIdx.x][d]; s += v * v; }
      ssq[threadIdx.x] = s;
    }
    __syncthreads();
    for (int jj = 0; jj < 64; ++jj) {
      const int cand = base + t + jj;
      if (cand == g) continue;
      float dot = 0.0f;
#pragma unroll
      for (int d = 0; d < F1; ++d) dot += f[d] * sf[jj][d];
      const float d2 = ssq[jj] - 2.0f * dot;
      if (d2 < bd[KNN - 1]) {
        int p = KNN - 1;
        while (p > 0 && bd[p - 1] > d2) { bd[p] = bd[p - 1]; bi[p] = bi[p - 1]; --p; }
        bd[p] = d2; bi[p] = cand;
      }
    }
    __syncthreads();
  }
  for (int k = 0; k < KNN; ++k) nbr2[g * KNN + k] = bi[k];
}

// ============================================================
// f32 WMMA GEMM: Z[M,N] = X[M,K] @ W[K,N] + bias, via V_WMMA_F32_16X16X4_F32.
// Block = (32,4): one 16-row tile; the 4 waves cover a 64-column panel.
// Both X (16 x KCH) and W (KCH x 64) tiles are staged in LDS with coalesced
// float4 (global_load_b128) loads and shared by all 4 waves; next chunk is
// prefetched (global_prefetch_b8). Requires M%16==0, N%64==0, K%KCH==0.
// All 128 threads active everywhere -> EXEC all-ones for every WMMA.
// Row strides padded to KCH+4 / 64+4 so 16-lane column reads hit distinct banks.
// ============================================================
template <int KCH>
__global__ void gemm_wmma_f32(const float* __restrict__ X, int Kdim,
                              const float* __restrict__ W, int Ncols,
                              const float* __restrict__ bias,
                              float* __restrict__ Z) {
  __shared__ __align__(16) float Xs[16][KCH + 4];
  __shared__ __align__(16) float Ws[KCH][64 + 4];
  const int lane = threadIdx.x;                     // 0..31
  const int wave = threadIdx.y;                     // 0..3
  const int rowBase  = blockIdx.x * 16;
  const int colBlock = blockIdx.y * 64;             // 64-col panel per block
  const int colBase  = colBlock + wave * 16;
  const int m  = lane & 15;
  const int kk = (lane >> 4) << 1;                  // 0 (lanes 0-15) or 2 (lanes 16-31)
  const int tid = wave * 32 + lane;                 // 0..127
  v8f acc = {};
  for (int k0 = 0; k0 < Kdim; k0 += KCH) {
    // stage X tile: 16 rows x KCH cols, one float4 per thread per step
    for (int e = tid; e < 16 * (KCH / 4); e += 128) {
      const int r = e / (KCH / 4), c4 = e % (KCH / 4);
      const float4 v = *(const float4*)(X + (size_t)(rowBase + r) * Kdim + k0 + c4 * 4);
      *(float4*)&Xs[r][c4 * 4] = v;
    }
    // stage W tile: KCH rows x 64 cols, one float4 per thread per step
    for (int e = tid; e < KCH * 16; e += 128) {
      const int r = e / 16, c4 = e % 16;
      const float4 v = *(const float4*)(W + (size_t)(k0 + r) * Ncols + colBlock + c4 * 4);
      *(float4*)&Ws[r][c4 * 4] = v;
    }
    __syncthreads();
    if (k0 + KCH < Kdim) {  // speculative prefetch of the next K-chunk
      __builtin_prefetch(X + (size_t)(rowBase + (tid & 15)) * Kdim + k0 + KCH, 0, 3);
      __builtin_prefetch(W + (size_t)(k0 + KCH + (tid >> 4)) * Ncols + colBlock, 0, 3);
    }
#pragma unroll
    for (int k2 = 0; k2 < KCH; k2 += 4) {
      v2f a, b;
      a.x = Xs[m][k2 + kk];                 // A: M=m, K=k0+k2+kk
      a.y = Xs[m][k2 + kk + 1];             //    K=k0+k2+kk+1
      b.x = Ws[k2 + kk][wave * 16 + m];     // B: K=k0+k2+kk,  N=colBase+m
      b.y = Ws[k2 + kk + 1][wave * 16 + m]; //    K=k0+k2+kk+1
      acc = __builtin_amdgcn_wmma_f32_16x16x4_f32(false, a, false, b,
                                                  (short)0, acc, false, false);
    }
    __syncthreads();
  }
  const float bv = bias ? bias[colBase + m] : 0.0f;
  const int rowOff = rowBase + ((lane >> 4) << 3);   // C/D: M = r (+8 for hi half-wave)
#pragma unroll
  for (int r = 0; r < 8; ++r)
    Z[(size_t)(rowOff + r) * Ncols + colBase + m] = acc[r] + bv;
}

// ============================================================
// Stage 5 (DDm, factored): per-cluster max/min over K of relu(zc[i]+zn[j]),
// plus per-cluster partial sums for BN statistics.
// ============================================================
__global__ void ddm_edge_kernel(const float* __restrict__ zc, const float* __restrict__ zn,
                                const int* __restrict__ nbr2, float* __restrict__ ymax,
                                float* __restrict__ ymin, float* __restrict__ psum,
                                float* __restrict__ psumsq) {
  const int i = blockIdx.x;      // cluster
  const int c = threadIdx.x;     // channel (128)
  const float zcv = zc[(size_t)i * C2 + c];
  float vmax = -3.0e38f, vmin = 3.0e38f, s = 0.0f, s2 = 0.0f;
  for (int k = 0; k < KNN; ++k) {
    const int j = nbr2[i * KNN + k];
    float y = zcv + zn[(size_t)j * C2 + c];
    y = fmaxf(y, 0.0f);
    vmax = fmaxf(vmax, y);
    vmin = fminf(vmin, y);
    s += y; s2 += y * y;
  }
  const size_t o = (size_t)i * C2 + c;
  ymax[o] = vmax; ymin[o] = vmin; psum[o] = s; psumsq[o] = s2;
}

// Deterministic per-channel reduction of BN stats (fixed stride order + fixed tree).
__global__ void ddm_stats_kernel(const float* __restrict__ psum, const float* __restrict__ psumsq,
                                 float* __restrict__ mean, float* __restrict__ invstd) {
  const int c = blockIdx.x;      // channel
  const int t = threadIdx.x;     // 256
  __shared__ float sh1[256], sh2[256];
  float s = 0.0f, s2 = 0.0f;
  for (int i = t; i < NC; i += 256) {
    s  += psum[(size_t)i * C2 + c];
    s2 += psumsq[(size_t)i * C2 + c];
  }
  sh1[t] = s; sh2[t] = s2;
  __syncthreads();
  for (int off = 128; off > 0; off >>= 1) {
    if (t < off) { sh1[t] += sh1[t + off]; sh2[t] += sh2[t + off]; }
    __syncthreads();
  }
  if (t == 0) {
    const float mn  = sh1[0] / (float)EDGES;
    const float var = sh2[0] / (float)EDGES - mn * mn;
    mean[c]   = mn;
    invstd[c] = rsqrtf(var + BN_EPS);
  }
}

// BN + max-over-K commute: per-channel affine with slope g*invstd; pick max/min by sign.
__global__ void ddm_bnmax_kernel(const float* __restrict__ ymax, const float* __restrict__ ymin,
                                 const float* __restrict__ mean, const float* __restrict__ invstd,
                                 const float* __restrict__ g, const float* __restrict__ bb,
                                 float* __restrict__ out) {
  const int idx = blockIdx.x * 256 + threadIdx.x;
  if (idx >= NC * C2) return;
  const int c = idx & (C2 - 1);
  const float sc = g[c] * invstd[c];
  const float v = (sc >= 0.0f) ? ymax[idx] : ymin[idx];
  out[idx] = sc * (v - mean[c]) + bb[c];
}

// ============================================================
// Head: per-batch segment-max of y1 (512 clusters), then relu.
// ============================================================
__global__ void segmax_relu_kernel(const float* __restrict__ y1, float* __restrict__ y1p) {
  const int b = blockIdx.x;                           // 16
  const int c = blockIdx.y * 256 + threadIdx.x;       // 1024
  float m = -3.0e38f;
  const float* p = y1 + (size_t)b * PC * 1024 + c;
  for (int r = 0; r < PC; ++r) m = fmaxf(m, p[(size_t)r * 1024]);
  y1p[b * 1024 + c] = fmaxf(m, 0.0f);
}

// BatchNorm over the 16 sample rows, in place. One thread per channel.
__global__ void bn_rows_kernel(float* __restrict__ x, const float* __restrict__ g,
                               const float* __restrict__ bb, int C) {
  const int c = threadIdx.x;
  if (c >= C) return;
  float s = 0.0f;
  for (int r = 0; r < Bsz; ++r) s += x[r * C + c];
  const float mn = s / (float)Bsz;
  float v = 0.0f;
  for (int r = 0; r < Bsz; ++r) { const float d = x[r * C + c] - mn; v += d * d; }
  v /= (float)Bsz;
  const float is = rsqrtf(v + BN_EPS);
  for (int r = 0; r < Bsz; ++r) x[r * C + c] = g[c] * (x[r * C + c] - mn) * is + bb[c];
}

// Small dense layer on the 16-row activations (optionally fused relu).
__global__ void dense_kernel(const float* __restrict__ x, const float* __restrict__ W,
                             const float* __restrict__ bias, float* __restrict__ out,
                             int InC, int OutC, int applyRelu) {
  const int idx = blockIdx.x * 256 + threadIdx.x;
  if (idx >= Bsz * OutC) return;
  const int r = idx / OutC, c = idx % OutC;
  float acc = bias[c];
  const float* xr = x + (size_t)r * InC;
  for (int i = 0; i < InC; ++i) acc += xr[i] * W[(size_t)i * OutC + c];
  if (applyRelu) acc = fmaxf(acc, 0.0f);
  out[idx] = acc;
}

// Final linear + log_softmax per sample (deterministic serial row reduce).
__global__ void head_final_kernel(const float* __restrict__ y3, const float* __restrict__ W,
                                  const float* __restrict__ bias, float* __restrict__ out) {
  const int r = blockIdx.x;    // sample
  const int c = threadIdx.x;   // 64 threads; first 40 active
  __shared__ float sl[NCLS];
  __shared__ float lse;
  if (c < NCLS) {
    float acc = bias[c];
    const float* xr = y3 + r * 265;
    for (int i = 0; i < 265; ++i) acc += xr[i] * W[i * NCLS + c];
    sl[c] = acc;
  }
  __syncthreads();
  if (c == 0) {
    float mx = sl[0];
    for (int i = 1; i < NCLS; ++i) mx = fmaxf(mx, sl[i]);
    float s = 0.0f;
    for (int i = 0; i < NCLS; ++i) s += expf(sl[i] - mx);
    lse = mx + logf(s);
  }
  __syncthreads();
  if (c < NCLS) out[r * NCLS + c] = sl[c] - lse;
}

// ============================================================
// Launch pipeline. Workspace need ~= 87 MB (carved below).
// ============================================================
extern "C" void kernel_launch(void* const* d_in, const int* in_sizes, int n_in,
                              void* d_out, int out_size, void* d_ws, size_t ws_size,
                              hipStream_t stream) {
  (void)in_sizes; (void)n_in; (void)out_size; (void)ws_size;
  const float* pos  = (const float*)d_in[0];
  const float* W_ds = (const float*)d_in[1];
  const float* b_ds = (const float*)d_in[2];
  const float* W2   = (const float*)d_in[3];
  const float* b2   = (const float*)d_in[4];
  const float* g2   = (const float*)d_in[5];
  const float* bb2  = (const float*)d_in[6];
  const float* W_n1 = (const float*)d_in[7];
  const float* b_n1 = (const float*)d_in[8];
  const float* g1   = (const float*)d_in[9];
  const float* bb1  = (const float*)d_in[10];
  const float* W_n2 = (const float*)d_in[11];
  const float* b_n2 = (const float*)d_in[12];
  const float* gn2  = (const float*)d_in[13];
  const float* bn2b = (const float*)d_in[14];
  const float* W_n3 = (const float*)d_in[15];
  const float* b_n3 = (const float*)d_in[16];
  const float* g3   = (const float*)d_in[17];
  const float* bb3  = (const float*)d_in[18];
  const float* W_n4 = (const float*)d_in[19];
  const float* b_n4 = (const float*)d_in[20];
  float* out = (float*)d_out;

  char* w = (char*)d_ws;
  auto carve = [&](size_t bytes) -> void* {
    void* p = (void*)w;
    w += (bytes + 255) & ~(size_t)255;
    return p;
  };
  int*   nbr1  = (int*)  carve((size_t)Npts * KNN * 4);
  float* fdd   = (float*)carve((size_t)Npts * FDIM * 4);
  float* fgra  = (float*)carve((size_t)NC * FDIM * 4);
  float* fnb   = (float*)carve((size_t)NC * F1 * 4);
  int*   nbr2  = (int*)  carve((size_t)NC * KNN * 4);
  float* zc    = (float*)carve((size_t)NC * C2 * 4);
  float* zn    = (float*)carve((size_t)NC * C2 * 4);
  float* ymax  = (float*)carve((size_t)NC * C2 * 4);
  float* ymin  = (float*)carve((size_t)NC * C2 * 4);
  float* psum  = (float*)carve((size_t)NC * C2 * 4);
  float* psq   = (float*)carve((size_t)NC * C2 * 4);
  float* mean2 = (float*)carve((size_t)C2 * 4);
  float* istd2 = (float*)carve((size_t)C2 * 4);
  float* fdd2  = (float*)carve((size_t)NC * C2 * 4);
  float* y1    = (float*)carve((size_t)NC * 1024 * 4);
  float* y1p   = (float*)carve((size_t)Bsz * 1024 * 4);
  float* y2    = (float*)carve((size_t)Bsz * 512 * 4);
  float* y3    = (float*)carve((size_t)Bsz * 265 * 4);

  // 1) kNN on positions
  knn1_kernel<<<Npts / 256, 256, 0, stream>>>(pos, nbr1);
  // 2) DS directional edge features + max over K
  ds_kernel<<<Npts, F1, 0, stream>>>(pos, nbr1, W_ds, b_ds, fdd);
  // 3) graclus pair pool + per-channel norms
  pool_kernel<<<(NC * F1 + 255) / 256, 256, 0, stream>>>(fdd, fgra, fnb);
  // 4) kNN on 64-dim feature norms
  knn2_kernel<<<NC / 256, 256, 0, stream>>>(fnb, nbr2);
  // 5) DDm factored GEMMs (WMMA f32): zc = F @ W2[:192] + b2, zn = F @ W2[192:]
  dim3 gblk(32, 4);
  gemm_wmma_f32<64><<<dim3(NC / 16, C2 / 64), gblk, 0, stream>>>(fgra, FDIM, W2, C2, b2, zc);
  gemm_wmma_f32<64><<<dim3(NC / 16, C2 / 64), gblk, 0, stream>>>(fgra, FDIM, W2 + FDIM * C2, C2,
                                                                 nullptr, zn);
  // 6) edge relu + stats + BN-commuted max-over-K
  ddm_edge_kernel<<<NC, C2, 0, stream>>>(zc, zn, nbr2, ymax, ymin, psum, psq);
  ddm_stats_kernel<<<C2, 256, 0, stream>>>(psum, psq, mean2, istd2);
  ddm_bnmax_kernel<<<(NC * C2) / 256, 256, 0, stream>>>(ymax, ymin, mean2, istd2, g2, bb2, fdd2);
  // 7) head GEMM (WMMA f32) + segment max + BN chain
  gemm_wmma_f32<64><<<dim3(NC / 16, 1024 / 64), gblk, 0, stream>>>(fdd2, C2, W_n1, 1024, b_n1, y1);
  segmax_relu_kernel<<<dim3(Bsz, 1024 / 256), 256, 0, stream>>>(y1, y1p);
  bn_rows_kernel<<<1, 1024, 0, stream>>>(y1p, g1, bb1, 1024);
  dense_kernel<<<(Bsz * 512 + 255) / 256, 256, 0, stream>>>(y1p, W_n2, b_n2, y2, 1024, 512, 1);
  bn_rows_kernel<<<1, 512, 0, stream>>>(y2, gn2, bn2b, 512);
  dense_kernel<<<(Bsz * 265 + 255) / 256, 256, 0, stream>>>(y2, W_n3, b_n3, y3, 512, 265, 1);
  bn_rows_kernel<<<1, 265, 0, stream>>>(y3, g3, bb3, 265);
  head_final_kernel<<<Bsz, 64, 0, stream>>>(y3, W_n4, b_n4, out);
}